// ChamferLoss_42898133352708
// MI455X (gfx1250) — compile-verified
//
#include <hip/hip_runtime.h>

typedef float v2f __attribute__((ext_vector_type(2)));
typedef float v8f __attribute__((ext_vector_type(8)));

#define NPTS   4096
#define BATCH  4
#define EPS_F  1e-8f

// One block = 256 threads = 8 wave32. Each wave owns a 16-row tile of "query"
// points, sweeps the full "reference" cloud (staged in LDS) in 16-col chunks.
// The entire squared-distance formula is folded into one WMMA per chunk:
//   A row m = [-2px, -2py, -2pz, 1]   (K=4, f32)
//   B col n = [ gx,   gy,   gz, g2]
//   C       = q2[m]           (loop-invariant accumulator init)
//   D = C + A*B = q2 + g2 - 2 p.g = d2   -- finished 16x16 distance tile.
// grid.x = BATCH * (NPTS/128)
__global__ __launch_bounds__(256)
void chamfer_dir_kernel(const float* __restrict__ Q,   // B x 3 x NPTS (queries)
                        const float* __restrict__ R,   // B x 3 x NPTS (references)
                        float* __restrict__ minOut)    // B*NPTS per-query min d2
{
    __shared__ float sref[3 * NPTS];   // interleaved xyz, 48 KB
    const int tilesPerBatch = NPTS / 128;
    const int b    = blockIdx.x / tilesPerBatch;
    const int tile = blockIdx.x % tilesPerBatch;
    const int tid  = threadIdx.x;
    const int lane = tid & 31;
    const int wave = tid >> 5;

    // Stage reference cloud into LDS, interleaving (B,3,N) -> xyz-per-point.
    const float* Rb = R + (size_t)b * 3 * NPTS;
    for (int i = tid; i < 3 * NPTS; i += 256) {
        int d = i >> 12;           // coord index 0..2
        int n = i & (NPTS - 1);    // point index
        sref[n * 3 + d] = Rb[i];
    }
    __syncthreads();

    const float* Qb  = Q + (size_t)b * 3 * NPTS;
    const int m0     = tile * 128 + wave * 16;
    const int half   = lane >> 4;      // 0: lanes 0-15, 1: lanes 16-31
    const int l15    = lane & 15;

    // A-matrix 16x4 f32 (ISA 7.12.2): lanes 0-15 hold K=0,1; lanes 16-31 K=2,3.
    // Row m = [-2px, -2py, -2pz, 1].
    const int mA = m0 + l15;
    const float ax = Qb[0 * NPTS + mA];
    const float ay = Qb[1 * NPTS + mA];
    const float az = Qb[2 * NPTS + mA];
    v2f A;
    A.x = half ? (-2.0f * az) : (-2.0f * ax);
    A.y = half ? 1.0f         : (-2.0f * ay);

    // Loop-invariant C accumulator: query squared norms for the 8 rows this
    // lane's C VGPRs cover (C layout: VGPR v, lane<16 -> M=v ; lane>=16 -> M=v+8).
    const int mBase = m0 + half * 8;
    v8f qc;
#pragma unroll
    for (int v = 0; v < 8; ++v) {
        int m = mBase + v;
        float x = Qb[0 * NPTS + m];
        float y = Qb[1 * NPTS + m];
        float z = Qb[2 * NPTS + m];
        qc[v] = x * x + y * y + z * z;
    }

    float rm[8];
#pragma unroll
    for (int v = 0; v < 8; ++v) rm[v] = 3.0e38f;

    for (int n0 = 0; n0 < NPTS; n0 += 16) {
        const int n = n0 + l15;
        const float gx = sref[n * 3 + 0];
        const float gy = sref[n * 3 + 1];
        const float gz = sref[n * 3 + 2];
        const float g2 = gx * gx + gy * gy + gz * gz;

        // B-matrix 4x16 f32: lanes 0-15 K=0,1; lanes 16-31 K=2,3.
        // Col n = [gx, gy, gz, g2].
        v2f Bm;
        Bm.x = half ? gz : gx;
        Bm.y = half ? g2 : gy;

        // D = qc + A*B = full squared-distance tile.
        v8f D = __builtin_amdgcn_wmma_f32_16x16x4_f32(
                /*neg_a=*/false, A, /*neg_b=*/false, Bm,
                /*c_mod=*/(short)0, qc, /*reuse_a=*/false, /*reuse_b=*/false);

#pragma unroll
        for (int v = 0; v < 8; ++v)
            rm[v] = fminf(rm[v], D[v]);
    }

    // Min across the 16 lanes of each half (xor masks < 16 stay within a half).
#pragma unroll
    for (int off = 8; off >= 1; off >>= 1) {
#pragma unroll
        for (int v = 0; v < 8; ++v) {
            float o = __shfl_xor(rm[v], off, 32);
            rm[v] = fminf(rm[v], o);
        }
    }

    if (l15 == 0) {
#pragma unroll
        for (int v = 0; v < 8; ++v)
            minOut[(size_t)b * NPTS + mBase + v] = rm[v];
    }
}

// Single-block reduction: out = sum(sqrt(min_d2 + eps)) / count
// (fwd and bwd halves are contiguous in the workspace; both means share the
//  same denominator B*NPTS, so one fused sum works).
__global__ __launch_bounds__(256)
void chamfer_reduce_kernel(const float* __restrict__ mins,
                           float* __restrict__ out,
                           int total, float invCount)
{
    __shared__ float ssum[256];
    float s = 0.0f;
    for (int i = threadIdx.x; i < total; i += 256)
        s += sqrtf(mins[i] + EPS_F);
    ssum[threadIdx.x] = s;
    __syncthreads();
    for (int stride = 128; stride > 0; stride >>= 1) {
        if (threadIdx.x < stride) ssum[threadIdx.x] += ssum[threadIdx.x + stride];
        __syncthreads();
    }
    if (threadIdx.x == 0) out[0] = ssum[0] * invCount;
}

extern "C" void kernel_launch(void* const* d_in, const int* in_sizes, int n_in,
                              void* d_out, int out_size, void* d_ws, size_t ws_size,
                              hipStream_t stream) {
    const float* p = (const float*)d_in[0];  // B x 3 x M predicted
    const float* g = (const float*)d_in[1];  // B x 3 x N ground truth
    float* fwd = (float*)d_ws;               // B*NPTS floats
    float* bwd = fwd + BATCH * NPTS;         // B*NPTS floats (contiguous)

    dim3 grid(BATCH * (NPTS / 128));
    chamfer_dir_kernel<<<grid, 256, 0, stream>>>(p, g, fwd);  // forward mins
    chamfer_dir_kernel<<<grid, 256, 0, stream>>>(g, p, bwd);  // backward mins
    chamfer_reduce_kernel<<<1, 256, 0, stream>>>(
        fwd, (float*)d_out, 2 * BATCH * NPTS, 1.0f / (float)(BATCH * NPTS));
}